// GLCAM_25451976196499
// MI455X (gfx1250) — compile-verified
//
#include <hip/hip_runtime.h>
#include <math.h>

typedef float v2f __attribute__((ext_vector_type(2)));
typedef float v8f __attribute__((ext_vector_type(8)));

#define NB  32
#define NC  256
#define NH  56
#define NW  56
#define NHW 3136
#define NCH 128
#define NCQ 64

// workspace layout (float offsets)
#define WS_X12   0            // 32*128      = 4096
#define WS_QSUM  4096         // 32*64       = 2048
#define WS_KSUM  6144         // 32*64       = 2048
#define WS_PROB  8192         // 32*64*64    = 131072
#define WS_VBUF  139264       // 32*64*3136  = 6422528 (K-pair interleaved)

__device__ __forceinline__ float gelu_f(float v) {
  return 0.5f * v * (1.0f + erff(v * 0.70710678118654752440f));
}

// ---------------- kernel 1: channel pooling -> x12 = mean*cx_w + cx_b ----------------
__global__ __launch_bounds__(256) void k_pool(const float* __restrict__ x,
                                              const float* __restrict__ cx_w,
                                              const float* __restrict__ cx_b,
                                              float* __restrict__ x12) {
  const int c = blockIdx.x, b = blockIdx.y, t = threadIdx.x;
  const float* p = x + ((size_t)b * NC + c) * NHW;
  float s = 0.f;
  for (int i = t; i < NHW; i += 256) s += p[i];
  __shared__ float red[256];
  red[t] = s;
  __syncthreads();
  for (int off = 128; off > 0; off >>= 1) {
    if (t < off) red[t] += red[t + off];
    __syncthreads();
  }
  if (t == 0) x12[b * NCH + c] = (red[0] * (1.0f / NHW)) * cx_w[c] + cx_b[c];
}

// ---------------- kernel 2: depthwise 3x3 + scale + GELU -> out ch [0,128) ----------------
__global__ __launch_bounds__(256) void k_dwconv(const float* __restrict__ x,
                                                const float* __restrict__ dw_w,
                                                const float* __restrict__ dw_b,
                                                const float* __restrict__ x12,
                                                float* __restrict__ out) {
  const int c = blockIdx.x, b = blockIdx.y, t = threadIdx.x;
  const float* in = x + ((size_t)b * NC + c) * NHW;
  float w[9];
#pragma unroll
  for (int i = 0; i < 9; ++i) w[i] = dw_w[c * 9 + i];
  const float bias  = dw_b[c];
  const float scale = x12[b * NCH + c];
  float* o = out + ((size_t)b * NC + c) * NHW;
  for (int p = t; p < NHW; p += 256) {
    int y = p / NW, xx = p - y * NW;
    float s = 0.f;
#pragma unroll
    for (int ky = 0; ky < 3; ++ky) {
      int yy = y + ky - 1;
      if (yy < 0 || yy >= NH) continue;
#pragma unroll
      for (int kx = 0; kx < 3; ++kx) {
        int xc = xx + kx - 1;
        if (xc < 0 || xc >= NW) continue;
        s += in[yy * NW + xc] * w[ky * 3 + kx];
      }
    }
    o[p] = gelu_f((s + bias) * scale);
  }
}

// ---------------- kernel 3: fused qkvl GEMM + per-head GEMM (WMMA fp32) ----------------
// grid (49, 32), block 512 (16 waves). Each block: batch b, 64 spatial positions.
// LDS tiles are K-pair interleaved: elem (k, n) at ((k>>1)*64 + n)*2 + (k&1),
// so a lane's B operand {row k, row k+1} is one aligned ds_load_b64.
// A operands are loaded from global in-loop (L2-resident weights; compiler pipelines).
__global__ __launch_bounds__(512) void k_qkvl(
    const float* __restrict__ x,
    const float* __restrict__ qkvl_w, const float* __restrict__ qkvl_b,
    const float* __restrict__ q_w, const float* __restrict__ q_b,
    const float* __restrict__ k_w, const float* __restrict__ k_b,
    const float* __restrict__ v_w, const float* __restrict__ v_b,
    const float* __restrict__ l_w, const float* __restrict__ l_b,
    float* __restrict__ out, float* __restrict__ vbuf,
    float* __restrict__ qsum, float* __restrict__ ksum) {
  __shared__ float smem[NC * 64];  // 64 KB: X tile [64 pairs][64][2], then Y tile [128 pairs][64][2]
  const int tid  = threadIdx.x;
  const int wave = tid >> 5;
  const int lane = tid & 31;
  const int h = lane >> 4;   // lane half: K = 2h + vgpr (A/B), M = r + 8h (C/D)
  const int m = lane & 15;   // A row / B col / C col
  const int b  = blockIdx.y;
  const int p0 = blockIdx.x * 64;

  const v8f zero8 = {0.f, 0.f, 0.f, 0.f, 0.f, 0.f, 0.f, 0.f};

  // GEMM2 operands for this wave (prefetch while GEMM1 runs)
  const int hd = wave >> 2;
  const int mt = wave & 3;
  const float* hw_w = (hd == 0) ? q_w : (hd == 1) ? k_w : (hd == 2) ? v_w : l_w;
  const float* hw_b = (hd == 0) ? q_b : (hd == 1) ? k_b : (hd == 2) ? v_b : l_b;
  const float* arow2 = hw_w + (size_t)(16 * mt + m) * NCQ;
  __builtin_prefetch(arow2, 0, 3);

  // ---- stage X2 tile into LDS, pair-interleaved ----
  const float* x2 = x + ((size_t)b * NC + NCH) * NHW;
  for (int i = tid; i < NCH * 64; i += 512) {
    int k = i >> 6, n = i & 63;
    smem[((k >> 1) * 64 + n) * 2 + (k & 1)] = x2[(size_t)k * NHW + p0 + n];
  }
  __syncthreads();

  // ---- GEMM1: wave w -> output channels [16w,16w+16), 4 N-subtiles ----
  v8f acc[4];
#pragma unroll
  for (int nt = 0; nt < 4; ++nt) acc[nt] = zero8;
  const float* arow = qkvl_w + (size_t)(16 * wave + m) * NCH;
#pragma unroll 8
  for (int kk4 = 0; kk4 < 32; ++kk4) {
    v2f av = *(const v2f*)(arow + 4 * kk4 + 2 * h);  // {W[row][4k+2h], W[row][4k+2h+1]}
    const int kr = 2 * kk4 + h;                      // K row-pair index
#pragma unroll
    for (int nt = 0; nt < 4; ++nt) {
      v2f bm = *(const v2f*)&smem[(kr * 64 + nt * 16 + m) * 2];
      acc[nt] = __builtin_amdgcn_wmma_f32_16x16x4_f32(false, av, false, bm,
                                                      (short)0, acc[nt], false, false);
    }
  }
  __syncthreads();  // everyone done reading the X tile

  // ---- bias + GELU, store Y tile (pair-interleaved) over the same LDS ----
#pragma unroll
  for (int nt = 0; nt < 4; ++nt) {
#pragma unroll
    for (int rp = 0; rp < 4; ++rp) {
      int cp = 8 * wave + 4 * h + rp;                // channel pair index (c = 2cp, 2cp+1)
      v2f yv;
      yv.x = gelu_f(acc[nt][2 * rp]     + qkvl_b[2 * cp]);
      yv.y = gelu_f(acc[nt][2 * rp + 1] + qkvl_b[2 * cp + 1]);
      *(v2f*)&smem[(cp * 64 + nt * 16 + m) * 2] = yv;
    }
  }
  __syncthreads();

  // ---- GEMM2: head hd, M-subtile mt ----
  v8f acc2[4];
#pragma unroll
  for (int nt = 0; nt < 4; ++nt) acc2[nt] = zero8;
#pragma unroll 8
  for (int kk4 = 0; kk4 < 16; ++kk4) {
    v2f av = *(const v2f*)(arow2 + 4 * kk4 + 2 * h);
    const int kr = 32 * hd + 2 * kk4 + h;            // global channel pair index
#pragma unroll
    for (int nt = 0; nt < 4; ++nt) {
      v2f bm = *(const v2f*)&smem[(kr * 64 + nt * 16 + m) * 2];
      acc2[nt] = __builtin_amdgcn_wmma_f32_16x16x4_f32(false, av, false, bm,
                                                       (short)0, acc2[nt], false, false);
    }
  }

  // ---- epilogue ----
  if (hd == 3) {  // lfeat -> out channels [128,192)
#pragma unroll
    for (int nt = 0; nt < 4; ++nt)
#pragma unroll
      for (int r = 0; r < 8; ++r) {
        int cch = 16 * mt + 8 * h + r;
        out[((size_t)b * NC + NCH + cch) * NHW + p0 + nt * 16 + m] =
            gelu_f(acc2[nt][r] + hw_b[cch]);
      }
  } else if (hd == 2) {  // v -> vbuf, K-pair interleaved: ((c>>1)*NHW + p)*2 + (c&1)
    float* Vb = vbuf + (size_t)b * NCQ * NHW;
#pragma unroll
    for (int nt = 0; nt < 4; ++nt) {
      int p = p0 + nt * 16 + m;
#pragma unroll
      for (int rp = 0; rp < 4; ++rp) {
        int cp = 8 * mt + 4 * h + rp;
        v2f vv;
        vv.x = gelu_f(acc2[nt][2 * rp]     + hw_b[2 * cp]);
        vv.y = gelu_f(acc2[nt][2 * rp + 1] + hw_b[2 * cp + 1]);
        *(v2f*)&Vb[((size_t)cp * NHW + p) * 2] = vv;
      }
    }
  } else {  // q (hd==0) / k (hd==1): spatial sums only
    float* sums = (hd == 0) ? qsum : ksum;
#pragma unroll
    for (int r = 0; r < 8; ++r) {
      int cch = 16 * mt + 8 * h + r;
      float s = 0.f;
#pragma unroll
      for (int nt = 0; nt < 4; ++nt) s += gelu_f(acc2[nt][r] + hw_b[cch]);
      // reduce over the 16 lanes of this half (all share the same channel cch)
#pragma unroll
      for (int off = 1; off < 16; off <<= 1) s += __shfl_xor(s, off, 16);
      if (m == 0) atomicAdd(&sums[b * NCQ + cch], s);
    }
  }
}

// ---------------- kernel 4: probs[b][c][i] = softmax_i(qp[i]*kp[c]) ----------------
__global__ __launch_bounds__(64) void k_softmax(const float* __restrict__ qsum,
                                                const float* __restrict__ ksum,
                                                float* __restrict__ probs) {
  const int b = blockIdx.x, t = threadIdx.x;
  __shared__ float qp[NCQ];
  qp[t] = qsum[b * NCQ + t] * (1.0f / NHW);
  __syncthreads();
  const float kc = ksum[b * NCQ + t] * (1.0f / NHW);
  float mx = -3.402823e38f;
  for (int i = 0; i < NCQ; ++i) mx = fmaxf(mx, qp[i] * kc);
  float* row = probs + (size_t)b * NCQ * NCQ + (size_t)t * NCQ;
  float z = 0.f;
  for (int i = 0; i < NCQ; ++i) {
    float e = expf(qp[i] * kc - mx);
    z += e;
    row[i] = e;
  }
  const float inv = 1.0f / z;
  for (int i = 0; i < NCQ; ++i) row[i] *= inv;
}

// ---------------- kernel 5: out[192:256) = probs(64x64) @ V(64xHW), WMMA fp32 ----------------
// grid (98, 32), block 256 (8 waves): wave -> (mt = w%4, ntile = bx*2 + w/4)
__global__ __launch_bounds__(256) void k_attn_v(const float* __restrict__ probs,
                                               const float* __restrict__ vbuf,
                                               float* __restrict__ out) {
  const int tid  = threadIdx.x;
  const int wave = tid >> 5, lane = tid & 31;
  const int h = lane >> 4, m = lane & 15;
  const int b  = blockIdx.y;
  const int mt = wave & 3;
  const int n0 = (blockIdx.x * 2 + (wave >> 2)) * 16;

  const float* arow = probs + (size_t)b * NCQ * NCQ + (size_t)(16 * mt + m) * NCQ;
  const float* V    = vbuf + (size_t)b * NCQ * NHW;

  v8f acc = {0.f, 0.f, 0.f, 0.f, 0.f, 0.f, 0.f, 0.f};
#pragma unroll 8
  for (int kk4 = 0; kk4 < 16; ++kk4) {
    v2f av = *(const v2f*)(arow + 4 * kk4 + 2 * h);
    const int cp = 2 * kk4 + h;                      // V channel-pair index
    v2f bm = *(const v2f*)&V[((size_t)cp * NHW + n0 + m) * 2];
    acc = __builtin_amdgcn_wmma_f32_16x16x4_f32(false, av, false, bm,
                                                (short)0, acc, false, false);
  }
#pragma unroll
  for (int r = 0; r < 8; ++r) {
    int cch = 192 + 16 * mt + 8 * h + r;
    out[((size_t)b * NC + cch) * NHW + n0 + m] = acc[r];
  }
}

extern "C" void kernel_launch(void* const* d_in, const int* in_sizes, int n_in,
                              void* d_out, int out_size, void* d_ws, size_t ws_size,
                              hipStream_t stream) {
  const float* x      = (const float*)d_in[0];
  const float* dw_w   = (const float*)d_in[1];
  const float* dw_b   = (const float*)d_in[2];
  const float* cx_w   = (const float*)d_in[3];
  const float* cx_b   = (const float*)d_in[4];
  const float* qkvl_w = (const float*)d_in[5];
  const float* qkvl_b = (const float*)d_in[6];
  const float* q_w    = (const float*)d_in[7];
  const float* q_b    = (const float*)d_in[8];
  const float* k_w    = (const float*)d_in[9];
  const float* k_b    = (const float*)d_in[10];
  const float* v_w    = (const float*)d_in[11];
  const float* v_b    = (const float*)d_in[12];
  const float* l_w    = (const float*)d_in[13];
  const float* l_b    = (const float*)d_in[14];
  float* out = (float*)d_out;

  float* ws    = (float*)d_ws;
  float* x12   = ws + WS_X12;
  float* qsum  = ws + WS_QSUM;
  float* ksum  = ws + WS_KSUM;
  float* probs = ws + WS_PROB;
  float* vbuf  = ws + WS_VBUF;

  // zero q/k pooled accumulators (contiguous region)
  hipMemsetAsync(qsum, 0, 2 * NB * NCQ * sizeof(float), stream);

  k_pool  <<<dim3(NCH, NB), 256, 0, stream>>>(x, cx_w, cx_b, x12);
  k_dwconv<<<dim3(NCH, NB), 256, 0, stream>>>(x, dw_w, dw_b, x12, out);
  k_qkvl  <<<dim3(49, NB), 512, 0, stream>>>(x, qkvl_w, qkvl_b,
                                             q_w, q_b, k_w, k_b, v_w, v_b, l_w, l_b,
                                             out, vbuf, qsum, ksum);
  k_softmax<<<NB, 64, 0, stream>>>(qsum, ksum, probs);
  k_attn_v <<<dim3(98, NB), 256, 0, stream>>>(probs, vbuf, out);
}